// FluidElite_63488206570165
// MI455X (gfx1250) — compile-verified
//
#include <hip/hip_runtime.h>
#include <math.h>

typedef __attribute__((ext_vector_type(2))) float v2f;
typedef __attribute__((ext_vector_type(8))) float v8f;

#define LDP    784            // padded 769 -> 784 (49*16)
#define SLAB1  (784 * 784)
#define NROWS  1568           // stacked 2 slabs (rows of B, col-major ld)
#define NCOLA  769
#define NPAIR  385
#define SWEEPS 8
#define BTM    64
#define BTN    128
#define BTK    16

// ---- gfx1250 async global->LDS staging (fallback-safe) ----
#if defined(__has_builtin)
#if __has_builtin(__builtin_amdgcn_global_load_async_to_lds_b128)
#define HAVE_ASYNC_LDS 1
#endif
#if __has_builtin(__builtin_amdgcn_s_wait_asynccnt)
#define HAVE_WAIT_ASYNC 1
#endif
#endif

#if defined(HAVE_ASYNC_LDS)
typedef int v4i __attribute__((vector_size(16)));
typedef __attribute__((address_space(1))) v4i gv4i;   // global int4
typedef __attribute__((address_space(3))) v4i lv4i;   // LDS int4
__device__ __forceinline__ void async_b128(const float* g, float* l) {
  __builtin_amdgcn_global_load_async_to_lds_b128((gv4i*)g, (lv4i*)l, 0, 0);
}
__device__ __forceinline__ void wait_async0() {
#if defined(HAVE_WAIT_ASYNC)
  __builtin_amdgcn_s_wait_asynccnt(0);
#else
  asm volatile("s_wait_asynccnt 0x0" ::: "memory");
#endif
}
#endif

// ---------------------------------------------------------------- utilities
__device__ __forceinline__ float gelu_f(float x) {
  const float k0 = 0.7978845608028654f;  // sqrt(2/pi), jax tanh-approx gelu
  const float k1 = 0.044715f;
  float x3 = x * x * x;
  return 0.5f * x * (1.0f + tanhf(k0 * (x + k1 * x3)));
}

__device__ __forceinline__ float wave_sum(float v) {
#pragma unroll
  for (int off = 16; off > 0; off >>= 1) v += __shfl_xor(v, off, 32);
  return v;
}

// ---------------------------------------------------------------- zero
__global__ void zero_buf(float* __restrict__ p, int n) {
  int i = blockIdx.x * blockDim.x + threadIdx.x;
  if (i < n) p[i] = 0.0f;
}

// ---------------------------------------------------------------- build s_i slabs
// transposed==0: dst = 2 row-major slabs 784x784 (GEMM B operand)
// transposed==1: dst = col-major stacked B (784 cols x 1568 rows, ld 1568)
__global__ void build_slabs(const float* __restrict__ ctx,
                            const float* __restrict__ cores_h,
                            const float* __restrict__ cores_x,
                            const int* __restrict__ token_bits,
                            float* __restrict__ dst, int l, int transposed) {
  int idx = blockIdx.x * blockDim.x + threadIdx.x;
  if (idx >= 2 * SLAB1) return;
  int j   = idx / SLAB1;
  int rem = idx - j * SLAB1;
  int row = rem / LDP;
  int col = rem - row * LDP;
  float val = 0.0f;
  if (row < 768 && col < 768) {
    float w0 = cores_h[l * 4 + j * 2 + 0];
    float w1 = cores_h[l * 4 + j * 2 + 1];
    const float* cb = ctx + (size_t)l * 768 * 2 * 768 + (size_t)row * 2 * 768 + col;
    val = gelu_f(w0 * __builtin_nontemporal_load(cb) +
                 w1 * __builtin_nontemporal_load(cb + 768));
  } else if (row == 768 && col == 768) {
    int bit = token_bits[l];
    val = gelu_f(cores_x[l * 4 + j * 2 + bit]);
  }
  if (transposed) dst[(size_t)col * NROWS + j * LDP + row] = val;
  else            dst[idx] = val;
}

// ---------------------------------------------------------------- identity init (Vt)
__global__ void init_identity(float* __restrict__ V) {
  int idx = blockIdx.x * blockDim.x + threadIdx.x;
  if (idx >= SLAB1) return;
  int r = idx / LDP, c = idx - r * LDP;
  V[idx] = (r == c && r < 770) ? 1.0f : 0.0f;
}

// ---------------------------------------------------------------- WMMA f32 GEMM
// Bt[col][z*784+row] = (A @ Bm[z])[row][col]   (store transposed into col-major B)
// 8 waves (2x4), each a 32x32 tile (4 accumulators) -> 16 v_wmma_f32_16x16x4_f32
// per 16-deep K panel.  Panels staged to LDS via GLOBAL_LOAD_ASYNC_TO_LDS_B128.
__global__ __launch_bounds__(256) void gemm_nn_wmma(const float* __restrict__ A,
                                                    const float* __restrict__ Bm,
                                                    float* __restrict__ Bt) {
  const float* Bp  = Bm + (size_t)blockIdx.z * SLAB1;
  const int   zrow = blockIdx.z * LDP;

  __shared__ float As[BTM][BTK + 4];   // 64 x 20  (rows 80B -> 16B aligned chunks)
  __shared__ float Bs[BTK][BTN + 4];   // 16 x 132 (rows 528B -> 16B aligned chunks)

  int tid  = threadIdx.x;
  int lane = tid & 31;
  int wave = tid >> 5;
  int wm   = wave >> 2;                // 0..1
  int wn   = wave & 3;                 // 0..3
  int m0   = blockIdx.x * BTM;
  int n0   = blockIdx.y * BTN;
  int lr   = lane & 15;
  int hl   = (lane < 16) ? 0 : 2;      // half-wave K offset (ISA A/B frag layout)

  v8f acc00 = {0.f,0.f,0.f,0.f,0.f,0.f,0.f,0.f};
  v8f acc01 = acc00, acc10 = acc00, acc11 = acc00;

  for (int k0 = 0; k0 < LDP; k0 += BTK) {
    // ---- A panel 64x16 : 256 chunks of 4 floats, one per thread
    {
      int r = tid >> 2, cq = (tid & 3) * 4;
      int gr = m0 + r;
      const float* gp = A + (size_t)gr * LDP + k0 + cq;
      if (gr < LDP) {
#if defined(HAVE_ASYNC_LDS)
        async_b128(gp, &As[r][cq]);
#else
        *(float4*)&As[r][cq] = *(const float4*)gp;
#endif
      } else {
        As[r][cq] = 0.f; As[r][cq+1] = 0.f; As[r][cq+2] = 0.f; As[r][cq+3] = 0.f;
      }
    }
    // ---- B panel 16x128 : 512 chunks of 4 floats, two per thread
#pragma unroll
    for (int i = 0; i < 2; ++i) {
      int e = tid + i * 256;
      int r = e >> 5, cq = (e & 31) * 4;
      int gc = n0 + cq;
      const float* gp = Bp + (size_t)(k0 + r) * LDP + gc;
      if (gc < LDP) {
#if defined(HAVE_ASYNC_LDS)
        async_b128(gp, &Bs[r][cq]);
#else
        *(float4*)&Bs[r][cq] = *(const float4*)gp;
#endif
      } else {
        Bs[r][cq] = 0.f; Bs[r][cq+1] = 0.f; Bs[r][cq+2] = 0.f; Bs[r][cq+3] = 0.f;
      }
    }
#if defined(HAVE_ASYNC_LDS)
    wait_async0();
#endif
    __syncthreads();
#pragma unroll
    for (int kk = 0; kk < BTK; kk += 4) {
      int ar0 = wm * 32 + lr, ar1 = ar0 + 16;
      int bc0 = wn * 32 + lr, bc1 = bc0 + 16;
      v2f a0, a1, b0, b1;
      a0.x = As[ar0][kk + hl]; a0.y = As[ar0][kk + hl + 1];
      a1.x = As[ar1][kk + hl]; a1.y = As[ar1][kk + hl + 1];
      b0.x = Bs[kk + hl][bc0]; b0.y = Bs[kk + hl + 1][bc0];
      b1.x = Bs[kk + hl][bc1]; b1.y = Bs[kk + hl + 1][bc1];
      acc00 = __builtin_amdgcn_wmma_f32_16x16x4_f32(false, a0, false, b0, (short)0, acc00, false, false);
      acc01 = __builtin_amdgcn_wmma_f32_16x16x4_f32(false, a0, false, b1, (short)0, acc01, false, false);
      acc10 = __builtin_amdgcn_wmma_f32_16x16x4_f32(false, a1, false, b0, (short)0, acc10, false, false);
      acc11 = __builtin_amdgcn_wmma_f32_16x16x4_f32(false, a1, false, b1, (short)0, acc11, false, false);
    }
    __syncthreads();
  }
  // store transposed: Bt[(cc)*1568 + zrow + rr]
#define STORE_TILE(ACC, RB, CB)                                        \
  {                                                                    \
    _Pragma("unroll") for (int r_ = 0; r_ < 8; ++r_) {                 \
      int rr = (RB) + r_ + ((lane < 16) ? 0 : 8);                      \
      int cc = (CB) + lr;                                              \
      if (rr < LDP && cc < LDP)                                        \
        Bt[(size_t)cc * NROWS + zrow + rr] = ACC[r_];                  \
    }                                                                  \
  }
  STORE_TILE(acc00, m0 + wm * 32,      n0 + wn * 32)
  STORE_TILE(acc01, m0 + wm * 32,      n0 + wn * 32 + 16)
  STORE_TILE(acc10, m0 + wm * 32 + 16, n0 + wn * 32)
  STORE_TILE(acc11, m0 + wm * 32 + 16, n0 + wn * 32 + 16)
#undef STORE_TILE
}

// ---------------------------------------------------------------- one-sided Jacobi SVD
// Bt: col-major (784 cols x 1568 rows, ld NROWS) -> column ops fully coalesced.
// Vt: col-major accumulated right-rotations (Vt[f][c] = V[c][f], ld LDP).
__global__ __launch_bounds__(1024) void jacobi_sweeps(float* __restrict__ Bt,
                                                      float* __restrict__ Vt) {
  __shared__ int   sp[NPAIR], sq[NPAIR];
  __shared__ float sc[NPAIR], ss[NPAIR];
  int tid  = threadIdx.x;
  int lane = tid & 31;
  int wave = tid >> 5;   // 32 waves

  for (int sw = 0; sw < SWEEPS; ++sw) {
    for (int r = 0; r < 769; ++r) {
      // phase 1: column dots + rotation angles (one pair per wave-iter)
      for (int k = wave; k < NPAIR; k += 32) {
        int posA = k, posB = 769 - k;               // tournament pairing, 770 slots
        int p = (posA == 0) ? 0 : 1 + ((posA - 1 + r) % 769);
        int q = 1 + ((posB - 1 + r) % 769);
        const float* cp = Bt + (size_t)p * NROWS;
        const float* cq = Bt + (size_t)q * NROWS;
        float al = 0.f, be = 0.f, ga = 0.f;
        for (int row = lane; row < NROWS; row += 32) {
          float bp = cp[row], bq = cq[row];
          al += bp * bp; be += bq * bq; ga += bp * bq;
        }
        al = wave_sum(al); be = wave_sum(be); ga = wave_sum(ga);
        if (lane == 0) {
          float c = 1.0f, s = 0.0f;
          if (ga != 0.0f && ga * ga > 1e-26f * al * be) {
            float tau = (al - be) / (2.0f * ga);
            float t = -copysignf(1.0f / (fabsf(tau) + sqrtf(1.0f + tau * tau)), tau);
            c = 1.0f / sqrtf(1.0f + t * t);
            s = t * c;
          }
          sp[k] = p; sq[k] = q; sc[k] = c; ss[k] = s;
        }
      }
      __syncthreads();
      // phase 2: apply to B columns (contiguous)
      for (int w = tid; w < NPAIR * NROWS; w += 1024) {
        int k = w / NROWS, row = w - k * NROWS;
        float s = ss[k];
        if (s != 0.0f) {
          float c = sc[k];
          float* cp = Bt + (size_t)sp[k] * NROWS;
          float* cq = Bt + (size_t)sq[k] * NROWS;
          float bp = cp[row], bq = cq[row];
          cp[row] = c * bp - s * bq;
          cq[row] = s * bp + c * bq;
        }
      }
      // and to V (rows p,q of Vt, contiguous over c = 0..769)
      for (int w = tid; w < NPAIR * 770; w += 1024) {
        int k = w / 770, cc = w - k * 770;
        float s = ss[k];
        if (s != 0.0f) {
          float c = sc[k];
          float* vp = Vt + (size_t)sp[k] * LDP;
          float* vq = Vt + (size_t)sq[k] * LDP;
          float a = vp[cc], b = vq[cc];
          vp[cc] = c * a - s * b;
          vq[cc] = s * a + c * b;
        }
      }
      __syncthreads();
    }
  }
}

// ---------------------------------------------------------------- finalize step
__global__ __launch_bounds__(1024) void jacobi_finalize(const float* __restrict__ Bt,
                                                        const float* __restrict__ Vt,
                                                        float* __restrict__ carry,
                                                        float* __restrict__ vec,
                                                        int is_last) {
  __shared__ float s2[NCOLA], csum[NCOLA];
  __shared__ int   rk[NCOLA];
  int tid  = threadIdx.x;
  int lane = tid & 31;
  int wave = tid >> 5;

  for (int f = wave; f < NCOLA; f += 32) {
    const float* cf = Bt + (size_t)f * NROWS;
    float a = 0.f, cs = 0.f;
    for (int row = lane; row < NROWS; row += 32) {
      float b = cf[row];
      a += b * b; cs += b;
    }
    a = wave_sum(a); cs = wave_sum(cs);
    if (lane == 0) { s2[f] = a; csum[f] = cs; }
  }
  __syncthreads();
  for (int f = tid; f < NCOLA; f += 1024) {
    float v = s2[f];
    int rr = 0;
    for (int g = 0; g < NCOLA; ++g) {
      float u = s2[g];
      rr += (u > v) || (u == v && g < f);
    }
    rk[f] = rr;
  }
  __syncthreads();
  if (!is_last) {
    // carry[rank, c] = S_f * Vh[f, c] = sqrt(s2[f]) * Vt[f][c]   (coalesced both sides)
    for (int w = tid; w < NCOLA * NCOLA; w += 1024) {
      int f = w / NCOLA, c = w - f * NCOLA;
      int rr = rk[f];
      if (rr < 768)
        carry[(size_t)rr * LDP + c] = sqrtf(s2[f]) * Vt[(size_t)f * LDP + c];
    }
  } else {
    for (int f = tid; f < NCOLA; f += 1024) {
      int rr = rk[f];
      if (rr < 768) {
        float sv = sqrtf(s2[f]);
        vec[rr] = (sv > 0.0f) ? csum[f] / (sv * 1536.0f) : 0.0f;
      }
    }
  }
}

// ---------------------------------------------------------------- head GEMV
__global__ __launch_bounds__(256) void head_gemv(const float* __restrict__ W,
                                                 const float* __restrict__ bias,
                                                 const float* __restrict__ vec,
                                                 float* __restrict__ out) {
  int lane = threadIdx.x & 31;
  int wave = threadIdx.x >> 5;
  int row  = blockIdx.x * 8 + wave;
  if (row >= 32000) return;
  const float* wr = W + (size_t)row * 768;
  float acc = 0.0f;
  for (int f = lane; f < 768; f += 32) {
    __builtin_prefetch(wr + f + 256, 0, 1);
    acc += __builtin_nontemporal_load(wr + f) * vec[f];
  }
  acc = wave_sum(acc);
  if (lane == 0) out[row] = acc + bias[row];
}

// ---------------------------------------------------------------- host orchestration
extern "C" void kernel_launch(void* const* d_in, const int* in_sizes, int n_in,
                              void* d_out, int out_size, void* d_ws, size_t ws_size,
                              hipStream_t stream) {
  (void)in_sizes; (void)n_in; (void)out_size; (void)ws_size;
  const float* ctx        = (const float*)d_in[0];
  const int*   token_bits = (const int*)d_in[1];
  const float* cores_h    = (const float*)d_in[2];
  const float* cores_x    = (const float*)d_in[3];
  const float* head_w     = (const float*)d_in[4];
  const float* head_b     = (const float*)d_in[5];
  float*       out        = (float*)d_out;

  float* w     = (float*)d_ws;
  float* slab  = w;                 // 2*SLAB1  row-major slabs (GEMM B operand)
  float* Bt    = w + 2 * SLAB1;     // 2*SLAB1  col-major stacked M (784 x 1568)
  float* Vt    = w + 4 * SLAB1;     // SLAB1    col-major V
  float* carry = w + 5 * SLAB1;     // SLAB1    (pad region must stay zero)
  float* vec   = w + 6 * SLAB1;     // 784

  zero_buf<<<(SLAB1 + 255) / 256, 256, 0, stream>>>(carry, SLAB1);

  dim3 gbuild((2 * SLAB1 + 255) / 256);
  dim3 ggemm((LDP + BTM - 1) / BTM, (LDP + BTN - 1) / BTN, 2);
  dim3 gid((SLAB1 + 255) / 256);

  for (int i = 0; i <= 8; ++i) {
    if (i == 0) {
      build_slabs<<<gbuild, 256, 0, stream>>>(ctx, cores_h, cores_x, token_bits, Bt, 0, 1);
    } else {
      build_slabs<<<gbuild, 256, 0, stream>>>(ctx, cores_h, cores_x, token_bits, slab, i, 0);
      gemm_nn_wmma<<<ggemm, 256, 0, stream>>>(carry, slab, Bt);  // Bt = (carry@S_j)^T
    }
    init_identity<<<gid, 256, 0, stream>>>(Vt);
    jacobi_sweeps<<<1, 1024, 0, stream>>>(Bt, Vt);
    jacobi_finalize<<<1, 1024, 0, stream>>>(Bt, Vt, carry, vec, (i == 8) ? 1 : 0);
  }
  head_gemv<<<4000, 256, 0, stream>>>(head_w, head_b, vec, out);
}